// NeuronLayer_78108275245264
// MI455X (gfx1250) — compile-verified
//
#include <hip/hip_runtime.h>
#include <math.h>

typedef __attribute__((ext_vector_type(2))) float v2f;
typedef __attribute__((ext_vector_type(8))) float v8f;

#define NPRE  1024
#define NPOST 256
#define INV_E 0.36787944117144233f
#define E_F   2.7182818284590452f

// -------- gfx1250 async global->LDS staging (ASYNCcnt path), if exposed ----
#if defined(__gfx1250__) &&                                          \
    __has_builtin(__builtin_amdgcn_global_load_async_to_lds_b128) && \
    __has_builtin(__builtin_amdgcn_s_wait_asynccnt)
#define USE_ASYNC_LDS 1
#else
#define USE_ASYNC_LDS 0
#endif

#if USE_ASYNC_LDS
// Builtin signature (from clang diagnostic): (int4 AS1*, int4 AS3*, i32, i32)
typedef int v4i __attribute__((vector_size(4 * sizeof(int))));
typedef __attribute__((address_space(1))) v4i* g4p;
typedef __attribute__((address_space(3))) v4i* l4p;
// Generic LDS pointer: low 32 bits are the LDS byte offset (aperture rules).
__device__ __forceinline__ l4p to_lds(void* p) {
  return (l4p)(unsigned int)(unsigned long long)p;
}
// Generic global pointer has the same integer value as its AS(1) form.
__device__ __forceinline__ g4p to_glob(const void* p) {
  return (g4p)(unsigned long long)p;
}
#endif

// ---------------------------------------------------------------------------
// Reference-faithful principal-branch Lambert W (12 Halley iterations, same
// clips and branch-point init as the Python lambertw0). noinline: this only
// runs on prefilter survivors, so keep a single copy instead of 8 unrolled
// ones (I$ is the scarce resource, not issue slots, for cold code).
// ---------------------------------------------------------------------------
__device__ __attribute__((noinline)) float lambertw0_dev(float z) {
  z = fminf(fmaxf(z, -INV_E + 1e-8f), -1e-30f);
  float w = (z < -0.2f) ? (-1.0f + sqrtf(2.0f * (1.0f + E_F * z)))
                        : z * (1.0f - z);
#pragma unroll
  for (int i = 0; i < 12; ++i) {
    float ew  = expf(w);
    float f   = w * ew - z;
    float wp1 = w + 1.0f;
    w = w - f / (ew * wp1 - (w + 2.0f) * f / (2.0f * wp1));
  }
  return w;
}

// ---------------------------------------------------------------------------
// Phase 1: per-batch bitonic sort of 1024 (spike, index) pairs in LDS.
// Lexicographic (key, idx) comparator reproduces jnp's *stable* argsort.
// ---------------------------------------------------------------------------
__global__ __launch_bounds__(256) void sort_spikes_kernel(
    const float* __restrict__ spikes, float* __restrict__ s_out,
    int* __restrict__ i_out) {
  __shared__ float key[NPRE];
  __shared__ int   val[NPRE];
  const int b = blockIdx.x, tid = threadIdx.x;
  for (int i = tid; i < NPRE; i += 256) {
    key[i] = spikes[b * NPRE + i];
    val[i] = i;
  }
  __syncthreads();
  for (int size = 2; size <= NPRE; size <<= 1) {
    for (int stride = size >> 1; stride > 0; stride >>= 1) {
      for (int t = tid; t < NPRE / 2; t += 256) {
        const int lo = (t / stride) * (stride << 1) + (t % stride);
        const int hi = lo + stride;
        const bool up = ((lo & size) == 0);
        const float ka = key[lo], kb = key[hi];
        const int   va = val[lo], vb = val[hi];
        const bool gt = (ka > kb) || (ka == kb && va > vb);  // lexicographic
        if (gt == up) {
          key[lo] = kb; key[hi] = ka;
          val[lo] = vb; val[hi] = va;
        }
      }
      __syncthreads();
    }
  }
  for (int i = tid; i < NPRE; i += 256) {
    s_out[b * NPRE + i] = key[i];
    i_out[b * NPRE + i] = val[i];
  }
}

// ---------------------------------------------------------------------------
// Phase 2: one wave32 per (batch, 16-column tile). Prefix scan of
//   a = cumsum(w*e^s), b = cumsum(w*s*e^s)
// done 16 sorted rows at a time via V_WMMA_F32_16X16X4_F32 with a
// lower-triangular ones A-matrix (tensor-core scan). Candidates evaluated
// with a cheap monotone prefilter; full Halley only on survivors, then the
// exact reference masks re-applied.
//
// WMMA operand layout assumptions (ISA 7.12.2):
//   A 16x4 : lanes 0-15 M=lane {V0:K=0, V1:K=1}; lanes 16-31 M=lane-16
//            {V0:K=2, V1:K=3}
//   B 4x16 : V0 = {row0 @ lanes0-15, row2 @ lanes16-31}; V1 = {row1, row3}
//   C 16x16: VGPR r = {M=r @ lanes0-15, M=r+8 @ lanes16-31}, N = lane&15
// ---------------------------------------------------------------------------
__global__ __launch_bounds__(32) void ettfs_kernel(
    const float* __restrict__ W, const float* __restrict__ Ssorted,
    const int* __restrict__ Isorted, float* __restrict__ out) {
  __shared__ __align__(16) float ls[NPRE];
  __shared__ __align__(16) int   li[NPRE];
  const int b    = blockIdx.y;
  const int n0   = blockIdx.x * 16;
  const int lane = threadIdx.x;

#if USE_ASYNC_LDS
  // Each lane DMAs 16B chunks straight into LDS (GLOBAL_LOAD_ASYNC_TO_LDS_B128,
  // ASYNCcnt); no VGPR round-trip for the 8KB staging.
  for (int i = lane * 4; i < NPRE; i += 32 * 4) {
    __builtin_amdgcn_global_load_async_to_lds_b128(
        to_glob(Ssorted + (size_t)b * NPRE + i), to_lds(&ls[i]), 0, 0);
    __builtin_amdgcn_global_load_async_to_lds_b128(
        to_glob(Isorted + (size_t)b * NPRE + i), to_lds(&li[i]), 0, 0);
  }
  __builtin_amdgcn_s_wait_asynccnt(0);
  __syncthreads();
#else
  for (int i = lane; i < NPRE; i += 32) {
    ls[i] = Ssorted[b * NPRE + i];
    li[i] = Isorted[b * NPRE + i];
  }
  __syncthreads();
#endif

  const int col = lane & 15;
  const int n   = n0 + col;
  const int hv  = (lane < 16) ? 0 : 2;  // B-matrix row offset inside a K-quad
  const int hm  = (lane < 16) ? 0 : 8;  // C-matrix row offset

  float carryA = 0.0f, carryB = 0.0f;
  float tmin = INFINITY;

  for (int c = 0; c < NPRE / 16; ++c) {
    const int base = c * 16;

    // Software prefetch of next chunk's gathered weight rows (L2-resident
    // weights; hides gather latency) -> global_prefetch_b8.
    if (c + 1 < NPRE / 16) {
#pragma unroll
      for (int k = 0; k < 4; ++k) {
        const int jn = base + 16 + 4 * k + hv;
        __builtin_prefetch(&W[(size_t)li[jn] * NPOST + n], 0, 1);
        __builtin_prefetch(&W[(size_t)li[jn + 1] * NPOST + n], 0, 1);
      }
    }

    v8f accA = {};  // within-chunk inclusive prefix of w*e
    v8f accB = {};  // within-chunk inclusive prefix of w*s*e
#pragma unroll
    for (int k = 0; k < 4; ++k) {
      // ---- B slice: rows 4k+hv, 4k+hv+1 of the chunk, column n ----
      const int j0 = base + 4 * k + hv;
      const int j1 = j0 + 1;
      const float s0 = ls[j0], s1 = ls[j1];
      const float e0 = expf(s0), e1 = expf(s1);
      const float x0 = W[(size_t)li[j0] * NPOST + n] * e0;
      const float x1 = W[(size_t)li[j1] * NPOST + n] * e1;
      v2f Bx = {x0, x1};
      v2f By = {x0 * s0, x1 * s1};
      // ---- A slice: lower-triangular ones, columns 4k+hv, 4k+hv+1 ----
      const int m   = col;          // output row M
      const int kc0 = 4 * k + hv;   // K index carried by V0
      v2f Av = {(kc0 <= m) ? 1.0f : 0.0f, (kc0 + 1 <= m) ? 1.0f : 0.0f};

      accA = __builtin_amdgcn_wmma_f32_16x16x4_f32(false, Av, false, Bx,
                                                   (short)0, accA, false, false);
      accB = __builtin_amdgcn_wmma_f32_16x16x4_f32(false, Av, false, By,
                                                   (short)0, accB, false, false);
    }

    // ---- Candidate evaluation for the 8 rows this lane owns in C ----
#pragma unroll
    for (int r = 0; r < 8; ++r) {
      const int   j  = base + r + hm;
      const float a  = accA[r] + carryA;
      const float bb = accB[r] + carryB;
      const float sj = ls[j];
      const bool  last  = (j == NPRE - 1);
      const float snext = last ? INFINITY : ls[j + 1];

      const float a_safe = (fabsf(a) > 1e-10f) ? a : 1e-10f;
      const float inv_a  = 1.0f / a_safe;           // one IEEE divide
      const float boa    = bb * inv_a;
      const float ratio  = fminf(fmaxf(boa, -30.0f), 30.0f);
      const float z      = -expf(ratio) * inv_a;    // c = (thr-leak)*g = 1
      const bool  valid  = (a > 0.0f) && (z >= -INV_E);

      if (valid) {
        // Monotone prefilter on the clipped z: t>=sj <=> W0(zc)<=u, and
        // (for non-last rows) t<=snext <=> W0(zc)>=v, with w*e^w monotone
        // increasing on [-1,inf). Slack keeps it strictly lenient; exact
        // reference masks are re-applied after the Halley solve.
        const float zc = fminf(fmaxf(z, -INV_E + 1e-8f), -1e-30f);
        const float u  = boa - sj;
        bool pass = (u >= -1.0f);
        if (pass) {
          const float ue = u * __expf(u);
          pass = (zc <= ue + 1e-5f * fabsf(ue) + 1e-12f);
        }
        if (pass && !last) {
          const float v = boa - snext;
          if (v > -1.0f) {
            const float ve = v * __expf(v);
            pass = (zc >= ve - 1e-5f * fabsf(ve) - 1e-12f);
          }
        }
        if (pass) {
          const float t = boa - lambertw0_dev(z);
          const bool rej = (t < sj) || ((!last) && (t > snext));
          if (!rej) tmin = fminf(tmin, t);
        }
      }
    }

    // Carry out: row 15 of the chunk prefix lives in VGPR7, lanes 16-31.
    carryA += __shfl(accA[7], 16 + col, 32);
    carryB += __shfl(accB[7], 16 + col, 32);
  }

  // Each column's partial min is split across lane pairs (l, l+16).
  tmin = fminf(tmin, __shfl_xor(tmin, 16, 32));
  if (lane < 16) out[b * NPOST + n] = tmin;
}

// ---------------------------------------------------------------------------
extern "C" void kernel_launch(void* const* d_in, const int* in_sizes, int n_in,
                              void* d_out, int out_size, void* d_ws,
                              size_t ws_size, hipStream_t stream) {
  const float* spikes  = (const float*)d_in[0];  // (B, NPRE) f32
  const float* weights = (const float*)d_in[1];  // (NPRE, NPOST) f32
  float* out = (float*)d_out;                    // (B, NPOST) f32

  const int B = in_sizes[0] / NPRE;  // 128

  float* s_sorted = (float*)d_ws;
  int*   i_sorted = (int*)((char*)d_ws + (size_t)B * NPRE * sizeof(float));

  sort_spikes_kernel<<<B, 256, 0, stream>>>(spikes, s_sorted, i_sorted);

  dim3 grid(NPOST / 16, B);
  ettfs_kernel<<<grid, 32, 0, stream>>>(weights, s_sorted, i_sorted, out);
}